// GCN_18476949308097
// MI455X (gfx1250) — compile-verified
//
#include <hip/hip_runtime.h>
#include <hip/hip_bf16.h>

// ---------------------------------------------------------------------------
// GCN 2-layer inference for MI455X (gfx1250, wave32).
//   h   = segsum( norm * (x@W1)[src] -> dst ) ; relu
//   out = segsum( norm * (relu(h)@W2)[src] -> dst )
// norm = dinv[src]*dinv[dst], dinv = rsqrt(1 + indegree)  (self loops folded)
// GEMMs: V_WMMA_F32_16X16X4_F32 (full fp32 matrix path), global->LDS staging
// via GLOBAL_LOAD_ASYNC_TO_LDS_B128 (ASYNCcnt) where no ALU is needed on the
// way in. Scatter: wave-per-edge coalesced gather + global_atomic_add_f32.
// ---------------------------------------------------------------------------

typedef __attribute__((ext_vector_type(2))) float v2f;
typedef __attribute__((ext_vector_type(8))) float v8f;

#define GCN_N      100000
#define GCN_INDIM  512
#define GCN_HID    256
#define GCN_OUT    64

// ---------------- degree / normalization ----------------------------------
__global__ void gcn_deg_init(float* __restrict__ deg, int n) {
    int i = blockIdx.x * blockDim.x + threadIdx.x;
    if (i < n) deg[i] = 1.0f;              // self loop
}

__global__ void gcn_deg_count(const int* __restrict__ dst, float* __restrict__ deg, int e) {
    int i = blockIdx.x * blockDim.x + threadIdx.x;
    if (i < e) unsafeAtomicAdd(&deg[dst[i]], 1.0f);
}

__global__ void gcn_rsqrt(float* __restrict__ deg, int n) {
    int i = blockIdx.x * blockDim.x + threadIdx.x;
    if (i < n) {
        float d = deg[i];
        deg[i] = (d > 0.0f) ? rsqrtf(d) : 0.0f;
    }
}

// ---------------- WMMA f32 GEMM:  C[M,N] = A[M,K] @ B[K,N] ----------------
// Block: 256 threads = 8 waves, each wave one 16x16 C tile, block tile 32x64.
// Requires M % BM == 0 (true here: 100000 % 32 == 0), K % KC == 0, N % BN == 0
// -> no bounds checks, EXEC stays all-ones (WMMA requirement).
template<int BM, int BN, int KC, bool RELU_A>
__global__ __launch_bounds__(256)
void gcn_wmma_gemm_f32(const float* __restrict__ A, const float* __restrict__ B,
                       float* __restrict__ C, int M, int K, int N) {
    // A stride 36 floats = 144B: rows 16B-aligned for async B128 stores;
    // banks 36*m mod 64 distinct for m=0..15, hi-half (+2) lands on a
    // disjoint (==2 mod 4) bank set -> conflict-free b64 fragment reads.
    __shared__ float As[BM][KC + 4];
    // B stride 80 floats = 320B: 16B-aligned, half-waves hit disjoint banks.
    __shared__ float Bs[KC][BN + 16];

    const int tid  = threadIdx.x;
    const int wave = tid >> 5;     // 0..7
    const int lane = tid & 31;
    const int lo   = lane & 15;    // N index within tile (A: M index)
    const int hi   = lane >> 4;    // selects K pair (A/B), M half (C)

    const int blockM = blockIdx.y * BM;
    const int blockN = blockIdx.x * BN;
    const int mBase  = (wave >> 2) * 16;  // 0 / 16
    const int nBase  = (wave & 3) * 16;   // 0 / 16 / 32 / 48

    v8f acc = {0.f, 0.f, 0.f, 0.f, 0.f, 0.f, 0.f, 0.f};

    for (int k0 = 0; k0 < K; k0 += KC) {
        // --- A tile: BM x KC (32x32 f32), one 16B chunk per thread --------
        {
            int r = tid >> 3;              // 0..31
            int c = (tid & 7) * 4;         // 0..28
            const float* gp = A + (size_t)(blockM + r) * K + (k0 + c);
            if (RELU_A) {
                // ReLU must touch the data -> register staging path
                float4 v = *(const float4*)gp;
                v.x = fmaxf(v.x, 0.f); v.y = fmaxf(v.y, 0.f);
                v.z = fmaxf(v.z, 0.f); v.w = fmaxf(v.w, 0.f);
                *(float4*)&As[r][c] = v;
            } else {
                unsigned lds = (unsigned)(unsigned long long)&As[r][c];
                asm volatile("global_load_async_to_lds_b128 %0, %1, off"
                             :: "v"(lds), "v"(gp) : "memory");
            }
        }
        // --- B tile: KC x BN (32x64 f32), two 16B chunks per thread -------
        {
            int c  = (tid & 15) * 4;       // 0..60
            int r0 = tid >> 4;             // 0..15
#pragma unroll
            for (int p = 0; p < 2; ++p) {
                int r = r0 + p * 16;
                const float* gp = B + (size_t)(k0 + r) * N + (blockN + c);
                unsigned lds = (unsigned)(unsigned long long)&Bs[r][c];
                asm volatile("global_load_async_to_lds_b128 %0, %1, off"
                             :: "v"(lds), "v"(gp) : "memory");
            }
        }
        // this wave's async copies landed in LDS, then cross-wave barrier
        asm volatile("s_wait_asynccnt 0x0" ::: "memory");
        __syncthreads();

        // --- 8 x V_WMMA_F32_16X16X4_F32 per chunk -------------------------
#pragma unroll
        for (int kk = 0; kk < KC; kk += 4) {
            // ISA 7.12.2 f32 layouts: lane(hi,lo) holds A[lo][kk+2*hi+j],
            // B[kk+2*hi+j][lo] in VGPR j.
            const float* ap = &As[mBase + lo][kk + 2 * hi];
            v2f af; af[0] = ap[0]; af[1] = ap[1];
            v2f bf;
            bf[0] = Bs[kk + 2 * hi + 0][nBase + lo];
            bf[1] = Bs[kk + 2 * hi + 1][nBase + lo];
            acc = __builtin_amdgcn_wmma_f32_16x16x4_f32(
                      /*neg_a=*/false, af, /*neg_b=*/false, bf,
                      /*c_mod=*/(short)0, acc,
                      /*reuse_a=*/false, /*reuse_b=*/false);
        }
        __syncthreads();   // LDS consumed by all waves before next chunk
    }

    // --- store C: VGPR v -> row mBase + 8*hi + v, col nBase + lo ----------
    const int col  = blockN + nBase + lo;
    const int row0 = blockM + mBase + 8 * hi;
#pragma unroll
    for (int v = 0; v < 8; ++v)
        C[(size_t)(row0 + v) * N + col] = acc[v];
}

// ---------------- self-loop init: out[i,:] = src[i,:] * dinv[i]^2 ---------
// launched with blockDim.x == D (256 or 64), one block per node
__global__ void gcn_selfloop_init(const float* __restrict__ src,
                                  const float* __restrict__ dinv,
                                  float* __restrict__ out, int D) {
    int row = blockIdx.x;
    float s = dinv[row];
    size_t idx = (size_t)row * D + threadIdx.x;
    out[idx] = src[idx] * (s * s);
}

// ---------------- edge scatter: out[dst] += h[src] * dinv[src]*dinv[dst] --
// one wave32 per edge; lanes stride the feature dim -> coalesced gather and
// coalesced global_atomic_add_f32 (both largely L2-resident: h/hw ~100MB each
// vs 192MB L2).
template<int D>
__global__ __launch_bounds__(256)
void gcn_edge_scatter(const float* __restrict__ h, const float* __restrict__ dinv,
                      const int* __restrict__ src, const int* __restrict__ dst,
                      float* __restrict__ out, int e) {
    int wid  = blockIdx.x * (blockDim.x >> 5) + (threadIdx.x >> 5);
    int lane = threadIdx.x & 31;
    if (wid >= e) return;
    int s = src[wid];
    int d = dst[wid];
    float w = dinv[s] * dinv[d];
    const float* hr  = h   + (size_t)s * D;
    float*       orr = out + (size_t)d * D;
#pragma unroll
    for (int c = lane; c < D; c += 32)
        unsafeAtomicAdd(&orr[c], hr[c] * w);
}

// ---------------------------------------------------------------------------
extern "C" void kernel_launch(void* const* d_in, const int* in_sizes, int n_in,
                              void* d_out, int out_size, void* d_ws, size_t ws_size,
                              hipStream_t stream) {
    const float* x  = (const float*)d_in[0];   // [N, 512]
    const float* W1 = (const float*)d_in[1];   // [512, 256]
    const float* W2 = (const float*)d_in[2];   // [256, 64]
    const int*   ei = (const int*)d_in[3];     // [2, E]: row0 = src, row1 = dst
    float* out = (float*)d_out;                // [N, 64]

    const int N = in_sizes[0] / GCN_INDIM;     // 100000 (multiple of 32)
    const int E = in_sizes[3] / 2;             // 3200000
    const int* src = ei;
    const int* dst = ei + E;

    // workspace layout (all fully overwritten every call):
    //   dinv [N] | hw [N,256] (reused as h2 [N,64]) | h [N,256]
    float* dinv = (float*)d_ws;
    float* hw   = dinv + N;
    float* h    = hw + (size_t)N * GCN_HID;

    const int T = 256;
    const int edgeBlocks = (E + 7) / 8;        // 8 waves (edges) per block

    // 1) degrees + dinv
    gcn_deg_init <<<(N + T - 1) / T, T, 0, stream>>>(dinv, N);
    gcn_deg_count<<<(E + T - 1) / T, T, 0, stream>>>(dst, dinv, E);
    gcn_rsqrt    <<<(N + T - 1) / T, T, 0, stream>>>(dinv, N);

    // 2) hw = x @ W1   (100000 x 512 x 256)
    {
        dim3 grid(GCN_HID / 64, N / 32);
        gcn_wmma_gemm_f32<32, 64, 32, false><<<grid, T, 0, stream>>>(
            x, W1, hw, N, GCN_INDIM, GCN_HID);
    }

    // 3) h = selfloop + scatter of hw  (D = 256)
    gcn_selfloop_init<<<N, GCN_HID, 0, stream>>>(hw, dinv, h, GCN_HID);
    gcn_edge_scatter<GCN_HID><<<edgeBlocks, T, 0, stream>>>(hw, dinv, src, dst, h, E);

    // 4) h2 = relu(h) @ W2   (100000 x 256 x 64), relu fused into A load;
    //    result written into hw's space (h2 = first N*64 floats of hw)
    {
        dim3 grid(GCN_OUT / 64, N / 32);
        gcn_wmma_gemm_f32<32, 64, 32, true><<<grid, T, 0, stream>>>(
            h, W2, hw, N, GCN_HID, GCN_OUT);
    }

    // 5) out = selfloop + scatter of h2  (D = 64)
    gcn_selfloop_init<<<N, GCN_OUT, 0, stream>>>(hw, dinv, out, GCN_OUT);
    gcn_edge_scatter<GCN_OUT><<<edgeBlocks, T, 0, stream>>>(hw, dinv, src, dst, out, E);
}